// LeafInstanceSegmentationModule_60876866453854
// MI455X (gfx1250) — compile-verified
//
#include <hip/hip_runtime.h>

typedef float v2f __attribute__((ext_vector_type(2)));
typedef float v8f __attribute__((ext_vector_type(8)));

#define NPTS   6144
#define BATCH  2
#define DEMB   32
#define R2     0.0009f      // 0.03^2
#define SIMTH  0.7f
#define EPSN   1e-8f

// ---------------------------------------------------------------- init ws ---
__global__ void lism_init(int* __restrict__ leafcnt) {
    leafcnt[0] = 0;
    leafcnt[1] = 0;
}

// ------------------------------------------------------------------- prep ---
// per point: |p|^2, normalized embedding row En = E / max(||E||, eps),
// per-batch leaf count
__global__ void lism_prep(const float* __restrict__ points,
                          const float* __restrict__ emb,
                          const int*   __restrict__ leaf,
                          float* __restrict__ pn,
                          float* __restrict__ en,
                          int*   __restrict__ leafcnt) {
    int idx = blockIdx.x * blockDim.x + threadIdx.x;
    if (idx >= BATCH * NPTS) return;
    float x = points[idx * 3 + 0];
    float y = points[idx * 3 + 1];
    float z = points[idx * 3 + 2];
    pn[idx] = x * x + y * y + z * z;
    float e[DEMB];
    float s = 0.0f;
    #pragma unroll
    for (int d = 0; d < DEMB; ++d) {
        e[d] = emb[idx * DEMB + d];
        s = fmaf(e[d], e[d], s);
    }
    float inv = 1.0f / fmaxf(sqrtf(s), EPSN);
    #pragma unroll
    for (int d = 0; d < DEMB; ++d)
        en[idx * DEMB + d] = e[d] * inv;
    atomicAdd(&leafcnt[idx / NPTS], leaf[idx]);
}

// ------------------------------------------------------------------- main ---
// one wave (32 threads) per 16-row i-tile; loops over all 16-col j-tiles.
__launch_bounds__(32)
__global__ void lism_main(const float* __restrict__ points,
                          const float* __restrict__ emb,
                          const int*   __restrict__ leaf,
                          const float* __restrict__ W1,
                          const float* __restrict__ b1,
                          const float* __restrict__ W2,
                          const float* __restrict__ b2,
                          const float* __restrict__ pn,
                          const float* __restrict__ en,
                          const int*   __restrict__ leafcnt,
                          float* __restrict__ out) {
    __shared__ float lds_mask[16 * 16];   // sim-mask tile, [i][j]
    __shared__ float lds_comb[16 * 64];   // [emb(32), mean_sim(32)] rows
    __shared__ float lds_h[16 * 32];      // hidden layer rows

    const int tile = blockIdx.x;              // 0 .. 2*384-1
    const int b    = tile / (NPTS / 16);
    const int ib   = (tile % (NPTS / 16)) * 16;
    const int l    = threadIdx.x;             // 0..31 (wave32)
    const int ln   = l & 15;
    const int hi   = l >> 4;                  // 0: lanes 0-15, 1: lanes 16-31
    const int off  = hi * 2;                  // K offset for A/B lane halves

    const float* Pb  = points + (size_t)b * NPTS * 3;
    const float* Eb  = emb    + (size_t)b * NPTS * DEMB;
    const float* Enb = en     + (size_t)b * NPTS * DEMB;
    const int*   Lb  = leaf   + (size_t)b * NPTS;
    const float* pnb = pn     + (size_t)b * NPTS;

    // ---- i-tile resident operands (A layouts: M = ln, K-half by lane half) ----
    const int irow = ib + ln;
    float px = Pb[irow * 3 + 0], py = Pb[irow * 3 + 1], pz = Pb[irow * 3 + 2];
    v2f a_p;                                  // points A: (x,y,z,0)
    a_p.x = hi ? pz : px;
    a_p.y = hi ? 0.0f : py;

    v2f a_e[8];                               // normalized-embedding A, 8 K-chunks
    #pragma unroll
    for (int c = 0; c < 8; ++c) {
        a_e[c].x = Enb[irow * DEMB + 4 * c + off];
        a_e[c].y = Enb[irow * DEMB + 4 * c + off + 1];
    }

    float pn_i[8];                            // pn for C-layout rows i = v + 8*hi
    #pragma unroll
    for (int v = 0; v < 8; ++v) pn_i[v] = pnb[ib + v + 8 * hi];

    v8f acc_lo = {};                          // sum_sim, d = ln
    v8f acc_hi = {};                          // sum_sim, d = ln + 16
    int cnt_nb[8]  = {0, 0, 0, 0, 0, 0, 0, 0};
    int cnt_sim[8] = {0, 0, 0, 0, 0, 0, 0, 0};

    // =========================== j loop ====================================
    for (int jb = 0; jb < NPTS; jb += 16) {
        const int jcol = jb + ln;
        // points B tile
        float qx = Pb[jcol * 3 + 0], qy = Pb[jcol * 3 + 1], qz = Pb[jcol * 3 + 2];
        v2f b_p;
        b_p.x = hi ? qz : qx;
        b_p.y = hi ? 0.0f : qy;
        float pn_j = pnb[jcol];
        int   lf_j = Lb[jcol];

        v8f zero = {};
        // point gram: 1 f32 WMMA (K=4, last K zero-padded)
        v8f gram = __builtin_amdgcn_wmma_f32_16x16x4_f32(
            false, a_p, false, b_p, (short)0, zero, false, false);

        // cosine sims: 8 chained f32 WMMAs over D=32 (pre-normalized rows)
        v8f sims = zero;
        #pragma unroll
        for (int c = 0; c < 8; ++c) {
            v2f b_e;
            b_e.x = Enb[jcol * DEMB + 4 * c + off];
            b_e.y = Enb[jcol * DEMB + 4 * c + off + 1];
            sims = __builtin_amdgcn_wmma_f32_16x16x4_f32(
                false, a_e[c], false, b_e, (short)0, sims, false, false);
        }

        // masks (C layout: i = v + 8*hi, j = ln); bounce mask through LDS
        #pragma unroll
        for (int v = 0; v < 8; ++v) {
            float d2 = fmaxf(pn_i[v] + pn_j - 2.0f * gram[v], 0.0f);
            bool nb = (d2 < R2) && (lf_j > 0);
            bool sm = nb && (sims[v] > SIMTH);
            cnt_nb[v]  += nb ? 1 : 0;
            cnt_sim[v] += sm ? 1 : 0;
            lds_mask[(v + 8 * hi) * 16 + ln] = sm ? 1.0f : 0.0f;
        }
        // (single wave: DS ops are in-order within a wave, no barrier needed)

        // sum_sim += mask(16x16) @ E_j(16x32): 4 K-chunks x 2 d-halves
        #pragma unroll
        for (int kk = 0; kk < 4; ++kk) {
            v2f a_m;
            a_m.x = lds_mask[ln * 16 + 4 * kk + off];
            a_m.y = lds_mask[ln * 16 + 4 * kk + off + 1];
            const int j0 = jb + 4 * kk + off;
            v2f b0, b1v;
            b0.x  = Eb[(j0 + 0) * DEMB + ln];
            b0.y  = Eb[(j0 + 1) * DEMB + ln];
            b1v.x = Eb[(j0 + 0) * DEMB + 16 + ln];
            b1v.y = Eb[(j0 + 1) * DEMB + 16 + ln];
            acc_lo = __builtin_amdgcn_wmma_f32_16x16x4_f32(
                false, a_m, false, b0, (short)0, acc_lo, false, false);
            acc_hi = __builtin_amdgcn_wmma_f32_16x16x4_f32(
                false, a_m, false, b1v, (short)0, acc_hi, false, false);
        }
    }

    // ---- reduce counts over j (within each 16-lane half) ----
    int cnb[8], csim[8];
    #pragma unroll
    for (int v = 0; v < 8; ++v) {
        int a = cnt_nb[v], s = cnt_sim[v];
        #pragma unroll
        for (int m = 1; m < 16; m <<= 1) {
            a += __shfl_xor(a, m, 32);
            s += __shfl_xor(s, m, 32);
        }
        cnb[v] = a;
        csim[v] = s;
    }

    // ---- mean_sim + stage [emb, mean] rows into LDS ----
    float e_lo[8], e_hi[8];
    #pragma unroll
    for (int v = 0; v < 8; ++v) {
        const int r = v + 8 * hi;
        const int i = ib + r;
        e_lo[v] = Eb[i * DEMB + ln];
        e_hi[v] = Eb[i * DEMB + 16 + ln];
        int den = csim[v] > 1 ? csim[v] : 1;
        float dnm = 1.0f / (float)den;
        lds_comb[r * 64 + ln]      = e_lo[v];
        lds_comb[r * 64 + 16 + ln] = e_hi[v];
        lds_comb[r * 64 + 32 + ln] = acc_lo[v] * dnm;
        lds_comb[r * 64 + 48 + ln] = acc_hi[v] * dnm;
    }

    // ---- MLP layer 1: h = relu(comb @ W1 + b1)  (W1: 64x32) ----
    float h_lo[8], h_hi[8];
    {
        float bl = b1[ln], bh = b1[16 + ln];
        #pragma unroll
        for (int v = 0; v < 8; ++v) { h_lo[v] = bl; h_hi[v] = bh; }
    }
    for (int k = 0; k < 64; ++k) {
        float wl = W1[k * DEMB + ln];
        float wh = W1[k * DEMB + 16 + ln];
        #pragma unroll
        for (int v = 0; v < 8; ++v) {
            float c = lds_comb[(v + 8 * hi) * 64 + k];
            h_lo[v] = fmaf(c, wl, h_lo[v]);
            h_hi[v] = fmaf(c, wh, h_hi[v]);
        }
    }
    #pragma unroll
    for (int v = 0; v < 8; ++v) {
        const int r = v + 8 * hi;
        h_lo[v] = fmaxf(h_lo[v], 0.0f);
        h_hi[v] = fmaxf(h_hi[v], 0.0f);
        lds_h[r * 32 + ln]      = h_lo[v];
        lds_h[r * 32 + 16 + ln] = h_hi[v];
    }

    // ---- MLP layer 2: out = h @ W2 + b2  (W2: 32x32) ----
    float o_lo[8], o_hi[8];
    {
        float bl = b2[ln], bh = b2[16 + ln];
        #pragma unroll
        for (int v = 0; v < 8; ++v) { o_lo[v] = bl; o_hi[v] = bh; }
    }
    for (int k = 0; k < 32; ++k) {
        float wl = W2[k * DEMB + ln];
        float wh = W2[k * DEMB + 16 + ln];
        #pragma unroll
        for (int v = 0; v < 8; ++v) {
            float hh = lds_h[(v + 8 * hi) * 32 + k];
            o_lo[v] = fmaf(hh, wl, o_lo[v]);
            o_hi[v] = fmaf(hh, wh, o_hi[v]);
        }
    }

    // ---- select + store ----
    const bool bok = leafcnt[b] >= 10;
    float* Ob = out + (size_t)b * NPTS * DEMB;
    #pragma unroll
    for (int v = 0; v < 8; ++v) {
        const int i = ib + v + 8 * hi;
        bool cond = bok && (Lb[i] > 0) && (cnb[v] > 1) && (csim[v] > 0);
        Ob[i * DEMB + ln]      = cond ? o_lo[v] : e_lo[v];
        Ob[i * DEMB + 16 + ln] = cond ? o_hi[v] : e_hi[v];
    }
}

// ----------------------------------------------------------------- launch ---
extern "C" void kernel_launch(void* const* d_in, const int* in_sizes, int n_in,
                              void* d_out, int out_size, void* d_ws, size_t ws_size,
                              hipStream_t stream) {
    const float* points = (const float*)d_in[0];
    const float* emb    = (const float*)d_in[1];
    const int*   leaf   = (const int*)d_in[2];
    const float* W1     = (const float*)d_in[3];
    const float* b1     = (const float*)d_in[4];
    const float* W2     = (const float*)d_in[5];
    const float* b2     = (const float*)d_in[6];
    float* out = (float*)d_out;

    // workspace: [0..7] int leafcnt[2]; pn[B*N]; En[B*N*32]
    int*   leafcnt = (int*)d_ws;
    float* pnw = (float*)((char*)d_ws + 16);
    float* enw = pnw + BATCH * NPTS;

    lism_init<<<1, 1, 0, stream>>>(leafcnt);

    const int npts = BATCH * NPTS;
    lism_prep<<<(npts + 255) / 256, 256, 0, stream>>>(points, emb, leaf,
                                                      pnw, enw, leafcnt);

    const int ntiles = BATCH * (NPTS / 16);   // 768 i-tiles, 1 wave each
    lism_main<<<ntiles, 32, 0, stream>>>(points, emb, leaf, W1, b1, W2, b2,
                                         pnw, enw, leafcnt, out);

    (void)in_sizes; (void)n_in; (void)out_size; (void)ws_size;
}